// Yolo_455266534072
// MI455X (gfx1250) — compile-verified
//
#include <hip/hip_runtime.h>
#include <cstdint>

// YOLO decode: [B, 3*85, G, G] -> rows [B, 3*G*G, 85] with sigmoid/exp/grid/anchor math.
// Memory-bound (~441 MB traffic, ~19us roofline @ 23.3 TB/s). Strategy:
//  - per block: one (batch, anchor, 169-position chunk) tile = 85 x 169 floats
//  - TDM async tensor load of the RAW tile into LDS (one tensor_load_to_lds per wave,
//    each wave waits its own TENSORcnt), then barrier
//  - compute phase reads LDS transposed (stride 169 words -> conflict-free, gcd(169,64)=1),
//    applies per-channel transform, writes fully-coalesced contiguous output.

typedef __attribute__((ext_vector_type(4))) unsigned int v4u;
typedef __attribute__((ext_vector_type(8))) int          v8i;
typedef __attribute__((ext_vector_type(4))) int          v4i;

constexpr int CHUNK = 169;          // positions per block tile
constexpr int NELEM = 85 * CHUNK;   // 14365 floats per tile (57,460 B LDS)

template<int G, int HEAD>
__global__ __launch_bounds__(256) void yolo_decode(const float* __restrict__ in,
                                                   float* __restrict__ out)
{
    constexpr int   GG       = G * G;
    constexpr int   NCHUNK   = GG / CHUNK;            // 1 (G=13) or 4 (G=26)
    constexpr float STRIDE   = (float)(416 / G);      // 32 or 16
    constexpr int   HEAD_OFF = (HEAD == 0) ? 0 : 3 * 169;  // 507 rows for head1
    constexpr int   TOTROWS  = 3 * 169 + 3 * 676;     // 2535

    __shared__ float lds[NELEM];

    const int t     = threadIdx.x;
    int       blk   = blockIdx.x;
    const int chunk = blk % NCHUNK;  blk /= NCHUNK;
    const int a     = blk % 3;
    const int b     = blk / 3;
    const int posBase = chunk * CHUNK;

    // element offset of (b, anchor a, channel 0, position posBase)
    const size_t inBase = ((size_t)b * 255 + (size_t)a * 85) * GG + (size_t)posBase;

#if __has_builtin(__builtin_amdgcn_tensor_load_to_lds) && __has_builtin(__builtin_amdgcn_s_wait_tensorcnt)
    {
        // Each wave DMAs an 11-row (last wave: 8-row) slice of the 85x169 tile.
        // All waves execute the tensor op (no divergent branch around it) and
        // wait on their own per-wave TENSORcnt.
        const int wid      = t >> 5;                       // wave id 0..7 (wave32)
        const int rowStart = wid * 11;
        const int rows     = (rowStart + 11 <= 85) ? 11 : (85 - rowStart);

        const uint64_t gaddr = (uint64_t)(uintptr_t)(in + inBase + (size_t)rowStart * GG);
        const uint32_t laddr = (uint32_t)(uintptr_t)(&lds[0])     // low 32 bits = LDS byte offset
                             + (uint32_t)(rowStart * CHUNK * 4);

        // ---- D# group 0 (128b): count=1 | lds_addr | global_addr[56:0] | type=2
        v4u g0;
        g0.x = 1u;                                         // count=1, is_restore=0, gather off
        g0.y = laddr;                                      // lds_addr (bytes)
        g0.z = (uint32_t)(gaddr & 0xFFFFFFFFull);          // global_addr[31:0]
        g0.w = (uint32_t)((gaddr >> 32) & 0x01FFFFFFull)   // global_addr[56:32]
             | (2u << 30);                                 // type = 2 ("image")

        // ---- D# group 1 (256b)
        v8i g1;
        g1[0] = (int)(2u << 16);                           // data_size = 2 -> 4 bytes; mask=0
        g1[1] = (int)((uint32_t)(GG & 0xFFFF) << 16);      // tensor_dim0[15:0] (atomic addr = 0)
        g1[2] = (int)((((uint32_t)GG >> 16) & 0xFFFFu)     // tensor_dim0[31:16]
                      | (255u << 16));                     // tensor_dim1[15:0] = 255
        g1[3] = (int)((uint32_t)CHUNK << 16);              // tile_dim0 = 169 (tensor_dim1 hi = 0)
        g1[4] = rows;                                      // tile_dim1 = rows, tile_dim2 = 0
        g1[5] = GG;                                        // tensor_dim0_stride[31:0] = G*G
        g1[6] = 0;                                         // stride hi / dim1_stride lo
        g1[7] = 0;                                         // dim1_stride hi

        v4i gz4 = {0, 0, 0, 0};                            // groups 2/3: 2D tensor, unused dims
        v8i gz8 = {0, 0, 0, 0, 0, 0, 0, 0};
        __builtin_amdgcn_tensor_load_to_lds(g0, g1, gz4, gz4, gz8, 0);
        __builtin_amdgcn_s_wait_tensorcnt(0);
    }
#else
    // Fallback: plain coalesced loads of the raw tile into LDS.
    for (int i = t; i < NELEM; i += 256) {
        const int c = i / CHUNK;
        const int p = i - c * CHUNK;
        lds[c * CHUNK + p] = in[inBase + (size_t)c * GG + p];
    }
#endif
    __syncthreads();

    // bw = exp(w) * (ANCHOR/stride) * stride = exp(w) * ANCHOR  (pixels)
    constexpr float AW[2][3] = {{81.f, 135.f, 344.f}, {10.f, 23.f, 37.f}};
    constexpr float AH[2][3] = {{82.f, 169.f, 319.f}, {14.f, 27.f, 58.f}};
    const float aw = AW[HEAD][a];
    const float ah = AH[HEAD][a];

    // This tile's 169 rows x 85 cols are contiguous in the output.
    const size_t outBase =
        (((size_t)b * TOTROWS) + HEAD_OFF + (size_t)a * GG + (size_t)posBase) * 85;

    for (int f = t; f < NELEM; f += 256) {
        const int p   = f / 85;            // position within chunk
        const int c   = f - p * 85;        // output channel 0..84
        const int pos = posBase + p;
        const int gy  = pos / G;
        const int gx  = pos - gy * G;

        const float v  = lds[c * CHUNK + p];     // lane stride 169 words: conflict-free
        const float e  = __expf(v);              // one v_exp_f32 serves sigmoid AND exp(w/h)
        const float sg = e * __builtin_amdgcn_rcpf(e + 1.0f);   // sigmoid(v) = e/(1+e)

        float r = sg;                                          // conf + 80 classes
        if (c == 0) r = (sg + (float)gx) * STRIDE;             // bx
        if (c == 1) r = (sg + (float)gy) * STRIDE;             // by
        if (c == 2) r = e * aw;                                // bw
        if (c == 3) r = e * ah;                                // bh

        out[outBase + f] = r;                                  // fully coalesced
    }
}

extern "C" void kernel_launch(void* const* d_in, const int* in_sizes, int n_in,
                              void* d_out, int out_size, void* d_ws, size_t ws_size,
                              hipStream_t stream)
{
    const float* feat13 = (const float*)d_in[0];   // [B, 255, 13, 13]
    const float* feat26 = (const float*)d_in[1];   // [B, 255, 26, 26]
    float*       out    = (float*)d_out;           // [B, 2535, 85]

    const int B = in_sizes[0] / (255 * 169);

    dim3 blk(256);
    // head 0: stride-32, anchors (3,4,5); 1 chunk per (b,a)
    yolo_decode<13, 0><<<dim3((unsigned)(B * 3 * 1)), blk, 0, stream>>>(feat13, out);
    // head 1: stride-16, anchors (0,1,2); 4 chunks per (b,a)
    yolo_decode<26, 1><<<dim3((unsigned)(B * 3 * 4)), blk, 0, stream>>>(feat26, out);
}